// StructureConsistencyLoss_28802050687626
// MI455X (gfx1250) — compile-verified
//
#include <hip/hip_runtime.h>
#include <hip/hip_bf16.h>

// ---------------------------------------------------------------------------
// StructureConsistencyLoss for MI455X (gfx1250, wave32)
//
// Memory-bound problem: 128 MiB streaming read dominates (~5.5us @ 23.3TB/s).
// Strategy:
//   K1 counts  : int atomics -> counts[512]
//   K2 phase1  : LDS segment accumulation (128KB LDS accumulator per WG,
//                exploiting CDNA5's 320KB/WGP), float4 coalesced loads,
//                ds_add_f32 atomics, non-atomic flush to per-chunk partials
//   K3 reduce  : partial[8][512][2048] -> sums[512][2048] (deterministic)
//   K4 phase2  : per-16-pid wave computes diff = mean0-mean1 and accumulates
//                the Gram matrix via V_WMMA_F32_16X16X4_F32; diagonal of the
//                16x16 f32 accumulator = per-pid sum of squared diffs.
//                (A and B register images coincide for the symmetric update.)
//   K5 phase3  : 256-thread reduction -> scalar loss
// ---------------------------------------------------------------------------

#define B_ROWS   16384
#define C_DIM    2048
#define P_PIDS   256
#define NSEG     512          // 2 * P_PIDS
#define CSLICE   64           // channels per workgroup slice
#define NSLICE   (C_DIM / CSLICE)   // 32
#define NCHUNK   8            // row chunks
#define CHUNK_ROWS (B_ROWS / NCHUNK) // 2048

typedef float v2f __attribute__((ext_vector_type(2)));
typedef float v8f __attribute__((ext_vector_type(8)));

#if defined(__has_builtin)
#if __has_builtin(__builtin_amdgcn_wmma_f32_16x16x4_f32)
#define HAVE_WMMA_F32X4 1
#endif
#endif

// ---------------------------------------------------------------- K1: counts
__global__ __launch_bounds__(256) void scl_counts_kernel(
    const int* __restrict__ pids, const int* __restrict__ camids,
    int* __restrict__ counts) {
  int i = blockIdx.x * 256 + threadIdx.x;
  if (i < B_ROWS) {
    int seg = pids[i] * 2 + camids[i];
    atomicAdd(&counts[seg], 1);
  }
}

// --------------------------------------------------------------- K2: phase 1
// grid: (NSLICE, NCHUNK), block: 256, dynamic LDS: 512*64*4 + 2048*4 bytes
__global__ __launch_bounds__(256) void scl_phase1_kernel(
    const float* __restrict__ f, const int* __restrict__ pids,
    const int* __restrict__ camids, float* __restrict__ partial) {
  extern __shared__ char smem[];
  float* lsum = (float*)smem;                          // [NSEG][CSLICE] 128KB
  int*   lseg = (int*)(smem + NSEG * CSLICE * 4);      // [CHUNK_ROWS]    8KB

  const int slice = blockIdx.x;       // channel slice
  const int chunk = blockIdx.y;       // row chunk
  const int tid   = threadIdx.x;
  const int rowStart = chunk * CHUNK_ROWS;

  // zero the LDS accumulator
  for (int i = tid; i < NSEG * CSLICE; i += 256) lsum[i] = 0.0f;
  // stage segment ids for this row chunk (avoids 16x redundant int loads)
  for (int i = tid; i < CHUNK_ROWS; i += 256) {
    int r = rowStart + i;
    lseg[i] = pids[r] * 2 + camids[r];
  }
  __syncthreads();

  const int tx = tid & 15;            // float4 column within slice
  const int ty = tid >> 4;            // row within tile of 16
  const int chBase = slice * CSLICE + tx * 4;

  for (int it = 0; it < CHUNK_ROWS; it += 16) {
    const int rr  = it + ty;
    const int seg = lseg[rr];
    const size_t gaddr = (size_t)(rowStart + rr) * C_DIM + chBase;
    // speculative prefetch of the next 16-row tile (global_prefetch_b8)
    if (it + 16 < CHUNK_ROWS)
      __builtin_prefetch(&f[gaddr + (size_t)16 * C_DIM], 0, 0);
    const float4 v = *(const float4*)&f[gaddr];
    float* dst = &lsum[seg * CSLICE + tx * 4];
    atomicAdd(dst + 0, v.x);          // ds_add_f32
    atomicAdd(dst + 1, v.y);
    atomicAdd(dst + 2, v.z);
    atomicAdd(dst + 3, v.w);
  }
  __syncthreads();

  // non-atomic flush to per-chunk partials (coalesced, deterministic tail)
  float* pbase = partial + (size_t)chunk * NSEG * C_DIM;
  for (int i = tid; i < NSEG * CSLICE; i += 256) {
    int seg = i / CSLICE;
    int ch  = i % CSLICE;
    pbase[(size_t)seg * C_DIM + slice * CSLICE + ch] = lsum[i];
  }
}

// ---------------------------------------------------------------- K3: reduce
__global__ __launch_bounds__(256) void scl_reduce_kernel(
    const float* __restrict__ partial, float* __restrict__ sums) {
  size_t idx = (size_t)blockIdx.x * 256 + threadIdx.x;  // 0 .. NSEG*C_DIM-1
  float s = 0.0f;
#pragma unroll
  for (int c = 0; c < NCHUNK; ++c)
    s += partial[(size_t)c * NSEG * C_DIM + idx];
  sums[idx] = s;
}

// --------------------------------------------------------------- K4: phase 2
// One wave32 per 16-pid group. Gram-diagonal via V_WMMA_F32_16X16X4_F32.
// A (16x4 f32) layout: lane L<16 holds rows M=L, K={0,1}; lane L+16 holds
// M=L-16, K={2,3}. B (4x16) mirrors it, so for the symmetric update D +=
// diff * diff^T the A and B register images are the same float2 per lane.
__global__ __launch_bounds__(32) void scl_phase2_kernel(
    const float* __restrict__ sums, const int* __restrict__ counts,
    float* __restrict__ mse, float* __restrict__ validf) {
  const int g    = blockIdx.x;        // pid group of 16
  const int lane = threadIdx.x;       // 0..31
  const int prow = g * 16 + (lane & 15);

  const int c0 = counts[2 * prow];
  const int c1 = counts[2 * prow + 1];
  const float inv0 = 1.0f / fmaxf((float)c0, 1.0f);
  const float inv1 = 1.0f / fmaxf((float)c1, 1.0f);
  const float* __restrict__ r0 = sums + (size_t)(2 * prow)     * C_DIM;
  const float* __restrict__ r1 = sums + (size_t)(2 * prow + 1) * C_DIM;
  const int chOff = (lane < 16) ? 0 : 2;

#ifdef HAVE_WMMA_F32X4
  v8f acc = {0.f, 0.f, 0.f, 0.f, 0.f, 0.f, 0.f, 0.f};
  for (int k = 0; k < C_DIM; k += 4) {
    const int ca = k + chOff;
    v2f ab;
    ab.x = r0[ca]     * inv0 - r1[ca]     * inv1;
    ab.y = r0[ca + 1] * inv0 - r1[ca + 1] * inv1;
    acc = __builtin_amdgcn_wmma_f32_16x16x4_f32(
        /*neg_a=*/false, ab, /*neg_b=*/false, ab,
        /*c_mod=*/(short)0, acc, /*reuse_a=*/false, /*reuse_b=*/false);
  }
  // Extract diagonal: D[i][i] lives in VGPR i, lane i (i<8) or
  // VGPR i-8, lane i+16 (i>=8).
  __shared__ float d[32][8];
#pragma unroll
  for (int j = 0; j < 8; ++j) d[lane][j] = acc[j];
  __syncthreads();
  if (lane < 16) {
    const int i = lane;
    const int srcLane = (i < 8) ? i : (i + 16);
    const int srcV    = (i < 8) ? i : (i - 8);
    const float ssq = d[srcLane][srcV];
    const bool valid = (c0 > 0) && (c1 > 0);
    mse[prow]    = valid ? (ssq * (1.0f / (float)C_DIM)) : 0.0f;
    validf[prow] = valid ? 1.0f : 0.0f;
  }
#else
  // VALU fallback: lanes L and L+16 cover disjoint channels of row prow.
  float ssq = 0.0f;
  for (int k = 0; k < C_DIM; k += 4) {
    const int ca = k + chOff;
    const float dx = r0[ca]     * inv0 - r1[ca]     * inv1;
    const float dy = r0[ca + 1] * inv0 - r1[ca + 1] * inv1;
    ssq += dx * dx + dy * dy;
  }
  __shared__ float d[32];
  d[lane] = ssq;
  __syncthreads();
  if (lane < 16) {
    const float tot = d[lane] + d[lane + 16];
    const bool valid = (c0 > 0) && (c1 > 0);
    mse[prow]    = valid ? (tot * (1.0f / (float)C_DIM)) : 0.0f;
    validf[prow] = valid ? 1.0f : 0.0f;
  }
#endif
}

// --------------------------------------------------------------- K5: phase 3
__global__ __launch_bounds__(256) void scl_phase3_kernel(
    const float* __restrict__ mse, const float* __restrict__ validf,
    float* __restrict__ out) {
  __shared__ float s1[256];
  __shared__ float s2[256];
  const int t = threadIdx.x;
  s1[t] = mse[t];
  s2[t] = validf[t];
  __syncthreads();
  for (int off = 128; off > 0; off >>= 1) {
    if (t < off) { s1[t] += s1[t + off]; s2[t] += s2[t + off]; }
    __syncthreads();
  }
  if (t == 0) {
    const float cnt  = s2[0];
    const float loss = s1[0] / fmaxf(cnt, 1.0f);
    out[0] = (cnt > 0.0f) ? loss : 0.0f;
  }
}

// ---------------------------------------------------------------------------
extern "C" void kernel_launch(void* const* d_in, const int* in_sizes, int n_in,
                              void* d_out, int out_size, void* d_ws, size_t ws_size,
                              hipStream_t stream) {
  const float* f      = (const float*)d_in[0];
  const int*   pids   = (const int*)d_in[1];
  const int*   camids = (const int*)d_in[2];
  float*       out    = (float*)d_out;

  // workspace layout
  char* ws = (char*)d_ws;
  float* partial = (float*)ws;                                   // 32 MB
  size_t off = (size_t)NCHUNK * NSEG * C_DIM * sizeof(float);
  float* sums = (float*)(ws + off);                              // 4 MB
  off += (size_t)NSEG * C_DIM * sizeof(float);
  int* counts = (int*)(ws + off);                                // 2 KB
  off += NSEG * sizeof(int);
  float* mse = (float*)(ws + off);                               // 1 KB
  off += P_PIDS * sizeof(float);
  float* validf = (float*)(ws + off);                            // 1 KB

  hipMemsetAsync(counts, 0, NSEG * sizeof(int), stream);

  scl_counts_kernel<<<(B_ROWS + 255) / 256, 256, 0, stream>>>(pids, camids, counts);

  const size_t p1_lds = (size_t)NSEG * CSLICE * 4 + (size_t)CHUNK_ROWS * 4; // 136 KB
  scl_phase1_kernel<<<dim3(NSLICE, NCHUNK), 256, p1_lds, stream>>>(
      f, pids, camids, partial);

  scl_reduce_kernel<<<(NSEG * C_DIM) / 256, 256, 0, stream>>>(partial, sums);

  scl_phase2_kernel<<<P_PIDS / 16, 32, 0, stream>>>(sums, counts, mse, validf);

  scl_phase3_kernel<<<1, 256, 0, stream>>>(mse, validf, out);
}